// CrossGG_88639535055055
// MI455X (gfx1250) — compile-verified
//
#include <hip/hip_runtime.h>
#include <hip/hip_bf16.h>

#define N_NODES 50000
#define N_EDGES 800000
#define NFEATS 128
#define NHID 128
#define NHEADS 8
#define DHEAD 16
#define ALPHA 0.2f
#define MTILES 5          // row tiles per wave; 50000 = 625 * (5*16)

typedef float v2f __attribute__((ext_vector_type(2)));
typedef float v8f __attribute__((ext_vector_type(8)));

// ---------------------------------------------------------------------------
// Kernel 1: Wh = h @ Wcat + Wb   via V_WMMA_F32_16X16X4_F32
// Block = 256 threads = 8 waves; wave w handles head w (cols w*16..w*16+15).
// Each wave computes MTILES=5 row tiles (M=80 per block), reusing one B
// fragment across 5 WMMAs per k-step.  Grid = 50000/80 = 625 (exact; EXEC
// stays all-ones through the WMMA loop as the ISA requires).
// ---------------------------------------------------------------------------
__global__ void gat_gemm_wh(const float* __restrict__ h,
                            const float* __restrict__ W,   // [8][128][16]
                            const float* __restrict__ Wb,  // [8][16] flat
                            float* __restrict__ Wh) {      // [N][128]
  const int wave = threadIdx.x >> 5;    // head / column tile
  const int lane = threadIdx.x & 31;
  const int half = lane >> 4;           // selects K pair within k-step
  const int lid  = lane & 15;           // M index (A), N index (B/C/D)
  const int row0 = blockIdx.x * (16 * MTILES);

  const float* __restrict__ Whead = W + wave * (NFEATS * DHEAD);

  const float* arow[MTILES];
#pragma unroll
  for (int tl = 0; tl < MTILES; ++tl)
    arow[tl] = h + (size_t)(row0 + tl * 16 + lid) * NFEATS;

  v8f acc[MTILES];
#pragma unroll
  for (int tl = 0; tl < MTILES; ++tl)
    acc[tl] = (v8f){0.f, 0.f, 0.f, 0.f, 0.f, 0.f, 0.f, 0.f};

#pragma unroll 2
  for (int k0 = 0; k0 < NFEATS; k0 += 4) {
    const int ka = k0 + half * 2;
    // B 4x16 f32: lane lid holds col N=lid, (b.x,b.y) = rows K=ka,ka+1
    v2f b;
    b.x = Whead[ka * DHEAD + lid];
    b.y = Whead[(ka + 1) * DHEAD + lid];
    // A 16x4 f32: lane lid holds row M=lid, (a.x,a.y) = K=ka,ka+1
#pragma unroll
    for (int tl = 0; tl < MTILES; ++tl) {
      v2f a = *(const v2f*)(arow[tl] + ka);      // 8B aligned (ka even)
      acc[tl] = __builtin_amdgcn_wmma_f32_16x16x4_f32(false, a, false, b,
                                                      (short)0, acc[tl],
                                                      false, false);
    }
  }

  const float bias = Wb[wave * DHEAD + lid];     // per-column bias
#pragma unroll
  for (int tl = 0; tl < MTILES; ++tl) {
#pragma unroll
    for (int r = 0; r < 8; ++r) {
      const int m = r + half * 8;                // C/D layout: VGPR r -> row
      Wh[(size_t)(row0 + tl * 16 + m) * NHID + wave * DHEAD + lid] =
          acc[tl][r] + bias;
    }
  }
}

// ---------------------------------------------------------------------------
// Kernel 2: per-node attention score halves. s_dst absorbs a_b.
// ---------------------------------------------------------------------------
__global__ void gat_scores(const float* __restrict__ Wh,
                           const float* __restrict__ a_src,
                           const float* __restrict__ a_dst,
                           const float* __restrict__ a_b,
                           float* __restrict__ s_src,
                           float* __restrict__ s_dst) {
  int t = blockIdx.x * blockDim.x + threadIdx.x;   // N * NHEADS
  if (t >= N_NODES * NHEADS) return;
  const int n = t >> 3, hh = t & 7;
  const float4* row = (const float4*)(Wh + (size_t)n * NHID + hh * DHEAD);
  const float4* as  = (const float4*)(a_src + hh * DHEAD);
  const float4* ad  = (const float4*)(a_dst + hh * DHEAD);
  float ss = 0.f, sd = 0.f;
#pragma unroll
  for (int q = 0; q < 4; ++q) {
    const float4 v = row[q], x = as[q], y = ad[q];
    ss += v.x * x.x + v.y * x.y + v.z * x.z + v.w * x.w;
    sd += v.x * y.x + v.y * y.y + v.z * y.z + v.w * y.w;
  }
  s_src[t] = ss;
  s_dst[t] = sd + a_b[hh];
}

// monotone float<->uint order encoding for atomic max
__device__ __forceinline__ unsigned enc_f32(float f) {
  unsigned u = __float_as_uint(f);
  return (u >> 31) ? ~u : (u | 0x80000000u);
}
__device__ __forceinline__ float dec_f32(unsigned u) {
  if (u == 0u) return 0.0f;  // segment with no edges
  return (u >> 31) ? __uint_as_float(u & 0x7FFFFFFFu) : __uint_as_float(~u);
}

// ---------------------------------------------------------------------------
// Kernel 3: edge logits + LeakyReLU, encoded atomic-max into m
// ---------------------------------------------------------------------------
__global__ void gat_edge_logits(const float* __restrict__ s_src,
                                const float* __restrict__ s_dst,
                                const int* __restrict__ esrc,
                                const int* __restrict__ edst,
                                float* __restrict__ ebuf,
                                unsigned* __restrict__ m_enc) {
  int t = blockIdx.x * blockDim.x + threadIdx.x;   // E * NHEADS
  if (t >= N_EDGES * NHEADS) return;
  const int e = t >> 3, hh = t & 7;
  const int s = esrc[e], d = edst[e];
  float x = s_src[s * NHEADS + hh] + s_dst[d * NHEADS + hh];
  x = (x >= 0.f) ? x : ALPHA * x;
  ebuf[t] = x;
  atomicMax(&m_enc[d * NHEADS + hh], enc_f32(x));
}

// ---------------------------------------------------------------------------
// Kernel 4: ex = exp(e - m[dst]); atomic-add into denom
// ---------------------------------------------------------------------------
__global__ void gat_edge_exp(const int* __restrict__ edst,
                             const unsigned* __restrict__ m_enc,
                             float* __restrict__ ebuf,
                             float* __restrict__ denom) {
  int t = blockIdx.x * blockDim.x + threadIdx.x;   // E * NHEADS
  if (t >= N_EDGES * NHEADS) return;
  const int e = t >> 3, hh = t & 7;
  const int d = edst[e];
  const float m = dec_f32(m_enc[d * NHEADS + hh]);
  const float ex = __expf(ebuf[t] - m);
  ebuf[t] = ex;
  atomicAdd(&denom[d * NHEADS + hh], ex);
}

// ---------------------------------------------------------------------------
// Kernel 5: out[dst] += (ex/denom[dst]) * Wh[src]
// One thread per (edge, 4-channel group): b128 gather + 4 f32 atomics.
// ---------------------------------------------------------------------------
__global__ void gat_edge_aggregate(const float* __restrict__ Wh,
                                   const float* __restrict__ ebuf,
                                   const float* __restrict__ denom,
                                   const int* __restrict__ esrc,
                                   const int* __restrict__ edst,
                                   float* __restrict__ out) {
  int t = blockIdx.x * blockDim.x + threadIdx.x;   // E * 32
  if (t >= N_EDGES * 32) return;
  const int e = t >> 5, q = t & 31;                // q: 4-channel group
  const int c0 = q << 2, hh = q >> 2;
  const int s = esrc[e], d = edst[e];
  const float ex = ebuf[e * NHEADS + hh];
  const float dn = denom[d * NHEADS + hh];
  const float attn = ex / fmaxf(dn, 1e-9f);
  const float4 w = *(const float4*)(Wh + (size_t)s * NHID + c0);  // b128
  float* o = out + (size_t)d * NHID + c0;
  atomicAdd(o + 0, attn * w.x);
  atomicAdd(o + 1, attn * w.y);
  atomicAdd(o + 2, attn * w.z);
  atomicAdd(o + 3, attn * w.w);
}

extern "C" void kernel_launch(void* const* d_in, const int* in_sizes, int n_in,
                              void* d_out, int out_size, void* d_ws, size_t ws_size,
                              hipStream_t stream) {
  const float* h      = (const float*)d_in[0];
  const int*   esrc   = (const int*)  d_in[1];
  const int*   edst   = (const int*)  d_in[2];
  const float* W      = (const float*)d_in[3];
  const float* Wb     = (const float*)d_in[4];
  const float* a_src  = (const float*)d_in[5];
  const float* a_dst  = (const float*)d_in[6];
  const float* a_b    = (const float*)d_in[7];
  float* out = (float*)d_out;

  // workspace layout (bytes) -- ~57.6 MB total
  char* ws = (char*)d_ws;
  size_t off = 0;
  float*    Wh     = (float*)(ws + off); off += (size_t)N_NODES * NHID * sizeof(float);
  float*    s_src  = (float*)(ws + off); off += (size_t)N_NODES * NHEADS * sizeof(float);
  float*    s_dstb = (float*)(ws + off); off += (size_t)N_NODES * NHEADS * sizeof(float);
  unsigned* m_enc  = (unsigned*)(ws + off); off += (size_t)N_NODES * NHEADS * sizeof(unsigned);
  float*    denom  = (float*)(ws + off); off += (size_t)N_NODES * NHEADS * sizeof(float);
  float*    ebuf   = (float*)(ws + off); off += (size_t)N_EDGES * NHEADS * sizeof(float);

  // init accumulators (d_out is poisoned by harness; atomics need zeros)
  hipMemsetAsync(out,   0, (size_t)N_NODES * NHID * sizeof(float), stream);
  hipMemsetAsync(m_enc, 0, (size_t)N_NODES * NHEADS * sizeof(unsigned), stream);
  hipMemsetAsync(denom, 0, (size_t)N_NODES * NHEADS * sizeof(float), stream);

  // 1) projection GEMM via f32 WMMA (625 blocks x 256 threads)
  gat_gemm_wh<<<N_NODES / (16 * MTILES), 256, 0, stream>>>(h, W, Wb, Wh);

  // 2) score halves
  {
    int n = N_NODES * NHEADS;
    gat_scores<<<(n + 255) / 256, 256, 0, stream>>>(Wh, a_src, a_dst, a_b, s_src, s_dstb);
  }

  // 3) edge logits + segment max
  {
    int n = N_EDGES * NHEADS;
    gat_edge_logits<<<(n + 255) / 256, 256, 0, stream>>>(s_src, s_dstb, esrc, edst, ebuf, m_enc);
  }

  // 4) exp + segment sum
  {
    int n = N_EDGES * NHEADS;
    gat_edge_exp<<<(n + 255) / 256, 256, 0, stream>>>(edst, m_enc, ebuf, denom);
  }

  // 5) weighted aggregation (E*32 threads, b128 gathers, f32 atomics)
  {
    int n = N_EDGES * 32;
    gat_edge_aggregate<<<(n + 255) / 256, 256, 0, stream>>>(Wh, ebuf, denom, esrc, edst, out);
  }
}